// Group_73495480369167
// MI455X (gfx1250) — compile-verified
//
#include <hip/hip_runtime.h>
#include <hip/hip_bf16.h>

#define Bn 16
#define Nn 8192
#define Gn 512
#define Mn 32
#define Dn 64

typedef __attribute__((ext_vector_type(2))) float v2f;
typedef __attribute__((ext_vector_type(8))) float v8f;

__device__ __forceinline__ unsigned short f32_to_bf16(float v) {
    unsigned u = __float_as_uint(v);
    u += 0x7FFFu + ((u >> 16) & 1u);   // round-to-nearest-even
    return (unsigned short)(u >> 16);
}
__device__ __forceinline__ float bf16_to_f32(unsigned short h) {
    return __uint_as_float(((unsigned)h) << 16);
}

// ---------------------------------------------------------------------------
// Kernel 1: Furthest Point Sampling. One workgroup (256 threads) per batch.
// xyz resident in LDS (SoA, 96KB); per-thread running min-dists in registers.
// ---------------------------------------------------------------------------
__global__ void fps_kernel(const float* __restrict__ xyz,
                           float* __restrict__ centers_ws,
                           float* __restrict__ out_center) {
    extern __shared__ unsigned char smem_raw[];
    float* xs   = (float*)smem_raw;
    float* ys   = xs + Nn;
    float* zs   = ys + Nn;
    float* bc   = zs + Nn;            // broadcast center coords (4 floats)
    float* wval = bc + 4;             // per-wave argmax value (8)
    int*   widx = (int*)(wval + 8);   // per-wave argmax index (8)

    const int b   = blockIdx.x;
    const int tid = threadIdx.x;
    const int lane = tid & 31;
    const int wid  = tid >> 5;
    const float* base = xyz + (size_t)b * Nn * 3;

    float dmin[32];
#pragma unroll
    for (int k = 0; k < 32; ++k) {
        int n = tid + (k << 8);
        float x = base[n * 3 + 0], y = base[n * 3 + 1], z = base[n * 3 + 2];
        xs[n] = x; ys[n] = y; zs[n] = z;
        dmin[k] = 1e10f;
    }
    __syncthreads();
    if (tid == 0) {  // first center is always point 0
        float x = xs[0], y = ys[0], z = zs[0];
        bc[0] = x; bc[1] = y; bc[2] = z;
        float* cw = centers_ws + (size_t)b * Gn * 3;
        cw[0] = x; cw[1] = y; cw[2] = z;
        float* co = out_center + (size_t)b * Gn * 3;
        co[0] = x; co[1] = y; co[2] = z;
    }
    __syncthreads();

    for (int g = 1; g < Gn; ++g) {
        float cx = bc[0], cy = bc[1], cz = bc[2];
        float lmax = -1.0f; int lidx = tid;
#pragma unroll
        for (int k = 0; k < 32; ++k) {
            int n = tid + (k << 8);
            float dx = xs[n] - cx, dy = ys[n] - cy, dz = zs[n] - cz;
            float d = dx * dx + dy * dy + dz * dz;
            d = fminf(dmin[k], d);
            dmin[k] = d;
            if (d > lmax) { lmax = d; lidx = n; }  // strict > keeps lowest n
        }
        // wave32 butterfly argmax, tie -> lower index (matches jnp.argmax)
        for (int off = 16; off > 0; off >>= 1) {
            float ov = __shfl_xor(lmax, off, 32);
            int   oi = __shfl_xor(lidx, off, 32);
            if (ov > lmax || (ov == lmax && oi < lidx)) { lmax = ov; lidx = oi; }
        }
        if (lane == 0) { wval[wid] = lmax; widx[wid] = lidx; }
        __syncthreads();
        if (tid == 0) {
            float bv = wval[0]; int bi = widx[0];
            for (int w = 1; w < 8; ++w) {
                float v = wval[w]; int i2 = widx[w];
                if (v > bv || (v == bv && i2 < bi)) { bv = v; bi = i2; }
            }
            float x = xs[bi], y = ys[bi], z = zs[bi];
            bc[0] = x; bc[1] = y; bc[2] = z;
            float* cw = centers_ws + ((size_t)(b * Gn) + g) * 3;
            cw[0] = x; cw[1] = y; cw[2] = z;
            float* co = out_center + ((size_t)(b * Gn) + g) * 3;
            co[0] = x; co[1] = y; co[2] = z;
        }
        __syncthreads();
    }
}

// ---------------------------------------------------------------------------
// Kernel 2: KNN. One workgroup (256 thr = 8 waves) per 16-query tile.
// Cross terms via V_WMMA_F32_16X16X4_F32 (K=3 padded to 4); the full
// 16 x 8192 distance tile lives in LDS as bf16 (256KB). Top-32 per query by
// iterative argmin with per-thread cached slice minima.
// ---------------------------------------------------------------------------
__global__ void knn_kernel(const float* __restrict__ xyz,
                           const float* __restrict__ centers,
                           int* __restrict__ knn_out) {
    extern __shared__ unsigned char smem_raw[];
    unsigned short* d2 = (unsigned short*)smem_raw;        // 16 * 8192 bf16
    float* qs   = (float*)(d2 + 16 * Nn);                  // 16 * 3
    float* qn   = qs + 16 * 3;                             // 16
    float* wval = qn + 16;                                 // 8
    int*   widx = (int*)(wval + 8);                        // 8
    int*   bsel = widx + 8;                                // 1

    const int wgid = blockIdx.x;
    const int b  = wgid >> 5;              // Gn/16 == 32 tiles per batch
    const int qb = (wgid & 31) << 4;       // first query of this tile
    const int tid  = threadIdx.x;
    const int lane = tid & 31;
    const int wid  = tid >> 5;
    const float* base = xyz + (size_t)b * Nn * 3;

    if (tid < 16) {
        const float* c = centers + ((size_t)(b * Gn) + qb + tid) * 3;
        float x = c[0], y = c[1], z = c[2];
        qs[tid * 3 + 0] = x; qs[tid * 3 + 1] = y; qs[tid * 3 + 2] = z;
        qn[tid] = x * x + y * y + z * z;
    }
    __syncthreads();

    // A-matrix (16x4 f32): lanes 0-15 hold K=0,1 ; lanes 16-31 hold K=2,3.
    const int khalf = lane >> 4;
    const int sub   = lane & 15;
    v2f A;
    A.x = khalf ? qs[sub * 3 + 2] : qs[sub * 3 + 0];   // K = 2*khalf
    A.y = khalf ? 0.0f            : qs[sub * 3 + 1];   // K = 2*khalf+1 (K=3 -> 0)

    // Each of 8 waves covers 64 column tiles of 16 ref points.
    for (int i = 0; i < 64; ++i) {
        int t = (wid << 6) + i;
        int n = (t << 4) + sub;
        const float* p = base + (size_t)n * 3;
        float rx = p[0], ry = p[1], rz = p[2];
        v2f Bm;
        Bm.x = khalf ? rz : rx;
        Bm.y = khalf ? 0.0f : ry;
        v8f acc = {};
        acc = __builtin_amdgcn_wmma_f32_16x16x4_f32(
            /*neg_a=*/false, A, /*neg_b=*/false, Bm,
            /*c_mod=*/(short)0, acc, /*reuse_a=*/false, /*reuse_b=*/false);
        float rn = rx * rx + ry * ry + rz * rz;
        // D layout: lanes 0-15: VGPR r -> M=r ; lanes 16-31: VGPR r -> M=8+r.
#pragma unroll
        for (int r = 0; r < 8; ++r) {
            int Mrow = r + (khalf << 3);
            float dd = qn[Mrow] + rn - 2.0f * acc[r];
            d2[Mrow * Nn + n] = f32_to_bf16(dd);
        }
    }
    __syncthreads();

    // Top-32 selection per query (queries processed sequentially).
    for (int m = 0; m < 16; ++m) {
        unsigned short* row = d2 + m * Nn;
        const int base_i = tid << 5;             // 32-element slice per thread
        float lval = 3.0e38f; int lidx = base_i;
        for (int j = 0; j < 32; ++j) {
            float v = bf16_to_f32(row[base_i + j]);
            if (v < lval) { lval = v; lidx = base_i + j; }
        }
        int* out = knn_out + ((size_t)(b * Gn) + qb + m) * Mn;
        for (int it = 0; it < Mn; ++it) {
            float rv = lval; int ri = lidx;
            for (int off = 16; off > 0; off >>= 1) {
                float ov = __shfl_xor(rv, off, 32);
                int   oi = __shfl_xor(ri, off, 32);
                if (ov < rv || (ov == rv && oi < ri)) { rv = ov; ri = oi; }
            }
            if (lane == 0) { wval[wid] = rv; widx[wid] = ri; }
            __syncthreads();
            if (tid == 0) {
                float bv = wval[0]; int bi = widx[0];
                for (int w = 1; w < 8; ++w) {
                    float v = wval[w]; int i2 = widx[w];
                    if (v < bv || (v == bv && i2 < bi)) { bv = v; bi = i2; }
                }
                bsel[0] = bi;
                out[it] = bi;
            }
            __syncthreads();
            int sel = bsel[0];
            if ((sel >> 5) == tid) {             // owner invalidates + rescans
                row[sel] = 0x7F80;               // +inf bf16
                lval = 3.0e38f; lidx = base_i;
                for (int j = 0; j < 32; ++j) {
                    float v = bf16_to_f32(row[base_i + j]);
                    if (v < lval) { lval = v; lidx = base_i + j; }
                }
            }
        }
        __syncthreads();
    }
}

// ---------------------------------------------------------------------------
// Kernel 3: gather + concat. One thread per new_points element; also emits
// the neighborhood tensor (shared first 3 channels).
// ---------------------------------------------------------------------------
__global__ void gather_kernel(const float* __restrict__ xyz,
                              const float* __restrict__ points,
                              const float* __restrict__ centers,
                              const int* __restrict__ knn,
                              float* __restrict__ out_neigh,
                              float* __restrict__ out_np) {
    const long long total = (long long)Bn * Gn * Mn * (Dn + 3);
    long long t = (long long)blockIdx.x * blockDim.x + threadIdx.x;
    if (t >= total) return;
    int c = (int)(t % (Dn + 3));
    long long r = t / (Dn + 3);
    int m = (int)(r % Mn);
    long long gq = r / Mn;                 // flat (b*Gn + g)
    int b = (int)(gq / Gn);
    int n = knn[gq * Mn + m];
    float v;
    if (c < 3) {
        v = xyz[((long long)b * Nn + n) * 3 + c] - centers[gq * 3 + c];
        out_neigh[(gq * Mn + m) * 3 + c] = v;
    } else {
        v = points[((long long)b * Nn + n) * Dn + (c - 3)];
    }
    out_np[t] = v;
}

extern "C" void kernel_launch(void* const* d_in, const int* in_sizes, int n_in,
                              void* d_out, int out_size, void* d_ws, size_t ws_size,
                              hipStream_t stream) {
    const float* xyz    = (const float*)d_in[0];
    const float* points = (const float*)d_in[1];

    float* out        = (float*)d_out;
    float* out_neigh  = out;                                         // [B,G,M,3]
    float* out_np     = out + (size_t)Bn * Gn * Mn * 3;              // [B,G,M,3+D]
    float* out_center = out_np + (size_t)Bn * Gn * Mn * (Dn + 3);    // [B,G,3]

    float* centers_ws = (float*)d_ws;                                // B*G*3 f32
    int*   knn_ws     = (int*)((char*)d_ws + sizeof(float) * Bn * Gn * 3);

    size_t fps_lds = sizeof(float) * (3 * Nn + 4 + 8) + sizeof(int) * 8;
    size_t knn_lds = sizeof(unsigned short) * 16 * Nn
                   + sizeof(float) * (16 * 3 + 16 + 8) + sizeof(int) * 9;

    // Opt into large dynamic LDS (320KB/WGP on CDNA5). Host-side config,
    // not a stream op -> safe under graph capture; deterministic each call.
    (void)hipFuncSetAttribute(reinterpret_cast<const void*>(fps_kernel),
                              hipFuncAttributeMaxDynamicSharedMemorySize, (int)fps_lds);
    (void)hipFuncSetAttribute(reinterpret_cast<const void*>(knn_kernel),
                              hipFuncAttributeMaxDynamicSharedMemorySize, (int)knn_lds);

    fps_kernel<<<Bn, 256, fps_lds, stream>>>(xyz, centers_ws, out_center);
    knn_kernel<<<Bn * (Gn / 16), 256, knn_lds, stream>>>(xyz, centers_ws, knn_ws);

    long long total = (long long)Bn * Gn * Mn * (Dn + 3);
    int blocks = (int)((total + 255) / 256);
    gather_kernel<<<blocks, 256, 0, stream>>>(xyz, points, centers_ws, knn_ws,
                                              out_neigh, out_np);
}